// _ProposalLayer_62088047231079
// MI455X (gfx1250) — compile-verified
//
#include <hip/hip_runtime.h>
#include <stdint.h>

#define NBOX   147456      // 9*128*128 total anchors
#define NTILE  (NBOX/16)   // 9216 WMMA tiles (16 boxes each)
#define NKEYS  262144      // pow2 pad for top-k sort
#define NPRE   6000        // PRE_NMS_TOP_N
#define NKEY2  8192        // pow2 pad for py2 sort
#define NPAD   6144        // chunk-padded proposal count (12 * 512)
#define NPOST  300         // POST_NMS_TOP_N
#define ROWW   192         // u32 words per suppression row (188 used)
#define SBLK   4096        // keys per LDS-resident bitonic block

typedef float v2f __attribute__((ext_vector_type(2)));
typedef float v8f __attribute__((ext_vector_type(8)));

// Monotone map: ascending uint <=> ascending float (handles -inf).
__device__ __forceinline__ unsigned fmono(float f){
  unsigned u = __float_as_uint(f);
  return (u & 0x80000000u) ? ~u : (u | 0x80000000u);
}

// ---------------- init: pads + zero output -------------------------------
__global__ void k_init(unsigned long long* keys, unsigned long long* key2,
                       float4* Pbox, float4* Pder, float* out)
{
  int t = blockIdx.x * blockDim.x + threadIdx.x;
  if (t < NKEYS - NBOX) keys[NBOX + t] = ~0ull;
  if (t < NKEY2 - NPRE) key2[NPRE + t] = ~0ull;
  if (t < NPAD - NPRE){
    Pbox[NPRE + t] = make_float4(0.f, 0.f, 0.f, 0.f);
    Pder[NPRE + t] = make_float4(0.f, 0.f, 0.f, 0.f);
  }
  if (t < NPOST * 4) out[t] = 0.0f;
}

// ------------- decode via WMMA: D = I(16x4) * deltas(4x16) + anchors -----
// One wave handles 16 boxes; exact under RNE (1.0*x + c).
__global__ void k_decode(const float* __restrict__ scores,
                         const float* __restrict__ deltas,
                         const float* __restrict__ anchors,
                         float4* __restrict__ boxes4,
                         unsigned long long* __restrict__ keys)
{
  const int lane = threadIdx.x & 31;
  const int tile = blockIdx.x * (blockDim.x >> 5) + (threadIdx.x >> 5);
  const int g = tile * 16 + (lane & 15);

  v2f b;
  {
    const float* dp = deltas + 4 * g;
    if (lane < 16){ b[0] = dp[0]; b[1] = dp[1]; }
    else          { b[0] = dp[2]; b[1] = dp[3]; }
  }
  v2f a; a[0] = 0.f; a[1] = 0.f;                 // A = 16x4 identity
  if (lane == 0 || lane == 18) a[0] = 1.0f;
  if (lane == 1 || lane == 19) a[1] = 1.0f;

  v8f c = {};
  if (lane < 16){
    const float* ap = anchors + 4 * g;
    c[0] = ap[0]; c[1] = ap[1]; c[2] = ap[2]; c[3] = ap[3];
  }
  v8f d = __builtin_amdgcn_wmma_f32_16x16x4_f32(
      false, a, false, b, (short)0, c, false, false);

  if (lane < 16){
    float x1 = fmaxf(d[0], 0.f);
    float y1 = fmaxf(d[1], 0.f);
    float bw = fmaxf(d[2], 0.f);
    float bh = fmaxf(d[3], 0.f);
    float x2 = fminf(x1 + bw - 1.0f, 127.0f);
    float y2 = fminf(y1 + bh - 1.0f, 127.0f);
    x1 = fminf(x1, 127.0f);
    y1 = fminf(y1, 127.0f);
    bw = x2 - x1 + 1.0f;
    bh = y2 - y1 + 1.0f;
    bool valid = (bw >= 16.0f) && (bh >= 16.0f);
    float sc = valid ? scores[g] : -__builtin_inff();
    boxes4[g] = make_float4(x1, y1, bw, bh);
    keys[g] = ((unsigned long long)(~fmono(sc)) << 32) | (unsigned)g;
  }
}

// ---------------- global bitonic compare-exchange step (j >= 4096) -------
__global__ void k_bitonic(unsigned long long* k, int j, int kk)
{
  int i = blockIdx.x * blockDim.x + threadIdx.x;
  int ixj = i ^ j;
  if (ixj > i){
    unsigned long long x = k[i], y = k[ixj];
    bool up = ((i & kk) == 0);
    if ((x > y) == up){ k[i] = y; k[ixj] = x; }
  }
}

// -------- fused LDS bitonic: all steps j = jhi..1 inside a 4096 block ----
// 32KB LDS per block; steps j<=2048 never cross a 4096-aligned boundary.
__global__ void k_bitonic_lds(unsigned long long* __restrict__ keys,
                              int kk, int jhi)
{
  __shared__ unsigned long long s[SBLK];
  const int base = blockIdx.x * SBLK;
  for (int t = threadIdx.x; t < SBLK; t += 256) s[t] = keys[base + t];
  __syncthreads();
  for (int j = jhi; j > 0; j >>= 1){
    for (int p = threadIdx.x; p < SBLK / 2; p += 256){
      int i   = ((p & ~(j - 1)) << 1) | (p & (j - 1));  // bit j clear
      int ixj = i | j;
      bool up = (((base + i) & kk) == 0);
      unsigned long long x = s[i], y = s[ixj];
      if ((x > y) == up){ s[i] = y; s[ixj] = x; }
    }
    __syncthreads();
  }
  for (int t = threadIdx.x; t < SBLK; t += 256) keys[base + t] = s[t];
}

// ------------- gather top-6000, build py2 sort keys ----------------------
__global__ void k_gather_top(const unsigned long long* __restrict__ keys,
                             const float4* __restrict__ boxes4,
                             float4* __restrict__ props,
                             unsigned long long* __restrict__ key2)
{
  int t = blockIdx.x * blockDim.x + threadIdx.x;
  if (t >= NPRE) return;
  unsigned g = (unsigned)keys[t];
  float4 b = boxes4[g];
  props[t] = b;
  float py2 = b.y + b.w - 1.0f;                 // y1 + bh - 1
  key2[t] = ((unsigned long long)(~fmono(py2)) << 32) | (unsigned)t;
}

// ------------- apply order, precompute x2/y2/area/valid ------------------
__global__ void k_gather_order(const unsigned long long* __restrict__ key2,
                               const float4* __restrict__ props,
                               float4* __restrict__ Pbox,
                               float4* __restrict__ Pder)
{
  int t = blockIdx.x * blockDim.x + threadIdx.x;
  if (t >= NPRE) return;
  unsigned s = (unsigned)key2[t];
  float4 p = props[s];
  Pbox[t] = p;
  float x2   = p.x + p.z - 1.0f;
  float y2   = p.y + p.w - 1.0f;
  float area = p.z * p.w;
  float vld  = (p.z >= 16.0f && p.w >= 16.0f) ? 1.0f : 0.0f;
  Pder[t] = make_float4(x2, y2, area, vld);
}

// ------------- suppression bitmask: row i, bit j>i if inter >= 0.7*area_j
// 8 waves/block = 8 rows/block. Column boxes double-buffered through LDS
// via async-to-LDS DMA: issue chunk c+1, s_wait_asynccnt 4 (chunk c's 4
// loads retired, next 4 still in flight), barrier, compute chunk c.
__global__ void k_suppress(const float4* __restrict__ Pbox,
                           const float4* __restrict__ Pder,
                           unsigned* __restrict__ suprow)
{
  __shared__ float4 ldsBox[2][512];
  __shared__ float4 ldsDer[2][512];
  const int wave = threadIdx.x >> 5;
  const int lane = threadIdx.x & 31;
  const int i = blockIdx.x * 8 + wave;          // row < 6000 (750*8 exact)

  float4 pb = Pbox[i];
  float4 pd = Pder[i];
  const float x1i = pb.x, y1i = pb.y, x2i = pd.x, y2i = pd.y;

  const unsigned long long boxAddr = (unsigned long long)(uintptr_t)Pbox;
  const unsigned long long derAddr = (unsigned long long)(uintptr_t)Pder;

  auto issue = [&](int c, int buf){
    unsigned bOff = (unsigned)(uintptr_t)&ldsBox[buf][0];
    unsigned dOff = (unsigned)(uintptr_t)&ldsDer[buf][0];
    for (int k = 0; k < 2; ++k){
      unsigned idx  = (unsigned)threadIdx.x + k * 256u;
      unsigned voff = (unsigned)(c * 512 + idx) * 16u;
      asm volatile("global_load_async_to_lds_b128 %0, %1, %2"
                   :: "v"(bOff + idx * 16u), "v"(voff), "s"(boxAddr)
                   : "memory");
      asm volatile("global_load_async_to_lds_b128 %0, %1, %2"
                   :: "v"(dOff + idx * 16u), "v"(voff), "s"(derAddr)
                   : "memory");
    }
  };

  issue(0, 0);                                  // prime the pipeline
  for (int c = 0; c < 12; ++c){
    const int buf = c & 1;
    if (c + 1 < 12){
      issue(c + 1, buf ^ 1);                    // prefetch next chunk
      asm volatile("s_wait_asynccnt 4" ::: "memory");   // chunk c retired
    } else {
      asm volatile("s_wait_asynccnt 0" ::: "memory");
    }
    __syncthreads();                            // all waves see chunk c

    const int j0 = c * 512;
    for (int it = 0; it < 16; ++it){
      int jj = it * 32 + lane;
      int j  = j0 + jj;
      float4 qb = ldsBox[buf][jj];
      float4 qd = ldsDer[buf][jj];
      float iw = fminf(x2i, qd.x) - fmaxf(x1i, qb.x) + 1.0f;
      float ih = fminf(y2i, qd.y) - fmaxf(y1i, qb.y) + 1.0f;
      iw = fmaxf(iw, 0.0f); ih = fmaxf(ih, 0.0f);
      bool bit = (j < NPRE) && (j > i) && (iw * ih >= 0.7f * qd.z);
      unsigned word = (unsigned)__ballot(bit);  // wave32: bit index == j&31
      if (lane == 0) suprow[(size_t)i * ROWW + (c * 16 + it)] = word;
    }
    __syncthreads();                            // buf free for reuse
  }
}

// ------------- sequential greedy scan (1 wave), emit up to 300 boxes -----
// Single-wave workgroup: barriers degrade to S_NOP on CDNA5.
__global__ void k_nms_scan(const float4* __restrict__ Pbox,
                           const float4* __restrict__ Pder,
                           const unsigned* __restrict__ suprow,
                           float* __restrict__ out)
{
  __shared__ unsigned removed[ROWW];
  __shared__ unsigned vbits[ROWW];
  const int lane = threadIdx.x;                 // 32 threads
  for (int w = lane; w < ROWW; w += 32) removed[w] = 0u;
  for (int it = 0; it < ROWW; ++it){            // prebuild validity bitmask
    int j = it * 32 + lane;
    bool v = (j < NPRE) && (Pder[j].w != 0.0f);
    unsigned word = (unsigned)__ballot(v);
    if (lane == 0) vbits[it] = word;
  }
  __syncthreads();

  int cnt = 0;
  for (int i = 0; i < NPRE; ++i){
    unsigned w = (unsigned)i >> 5, b = (unsigned)i & 31;
    bool keep = ((vbits[w] >> b) & 1u) && !((removed[w] >> b) & 1u);
    if (keep){                                  // uniform across the wave
      const unsigned* row = suprow + (size_t)i * ROWW;
      for (int t = lane; t < 188; t += 32) removed[t] |= row[t];
      if (lane == 0 && cnt < NPOST){
        float4 p = Pbox[i];
        out[cnt*4+0] = p.x; out[cnt*4+1] = p.y;
        out[cnt*4+2] = p.z; out[cnt*4+3] = p.w;
      }
      ++cnt;
    }
    __syncthreads();
  }
}

extern "C" void kernel_launch(void* const* d_in, const int* in_sizes, int n_in,
                              void* d_out, int out_size, void* d_ws, size_t ws_size,
                              hipStream_t stream)
{
  (void)in_sizes; (void)n_in; (void)out_size; (void)ws_size;
  const float* scores  = (const float*)d_in[0];   // (1,9,128,128) f32
  const float* deltas  = (const float*)d_in[1];   // (1,36,128,128) f32
  const float* anchors = (const float*)d_in[2];   // (9,128,128,4) f32
  float* out = (float*)d_out;                     // (1,300,4) f32

  char* ws = (char*)d_ws;                         // ~9.43 MB used
  unsigned long long* keys = (unsigned long long*)(ws);            // 2 MB
  float4* boxes4           = (float4*)(ws + 2097152);              // 2.36 MB
  float4* props            = (float4*)(ws + 4456448);              // 96 KB
  unsigned long long* key2 = (unsigned long long*)(ws + 4552448);  // 64 KB
  float4* Pbox             = (float4*)(ws + 4617984);              // 96 KB (padded)
  float4* Pder             = (float4*)(ws + 4716288);              // 96 KB (padded)
  unsigned* suprow         = (unsigned*)(ws + 4814592);            // 4.61 MB

  k_init<<<448, 256, 0, stream>>>(keys, key2, Pbox, Pder, out);
  k_decode<<<NTILE/8, 256, 0, stream>>>(scores, deltas, anchors, boxes4, keys);

  // full ascending bitonic sort of 262144 keys:
  // cross-block steps (j>=4096) in global memory, the rest fused in LDS.
  for (int kk = 2; kk <= NKEYS; kk <<= 1){
    int j = kk >> 1;
    for (; j >= SBLK; j >>= 1)
      k_bitonic<<<NKEYS/256, 256, 0, stream>>>(keys, j, kk);
    k_bitonic_lds<<<NKEYS/SBLK, 256, 0, stream>>>(keys, kk, j);
  }

  k_gather_top<<<24, 256, 0, stream>>>(keys, boxes4, props, key2);

  for (int kk = 2; kk <= NKEY2; kk <<= 1){
    int j = kk >> 1;
    for (; j >= SBLK; j >>= 1)
      k_bitonic<<<NKEY2/256, 256, 0, stream>>>(key2, j, kk);
    k_bitonic_lds<<<NKEY2/SBLK, 256, 0, stream>>>(key2, kk, j);
  }

  k_gather_order<<<24, 256, 0, stream>>>(key2, props, Pbox, Pder);
  k_suppress<<<NPRE/8, 256, 0, stream>>>(Pbox, Pder, suprow);
  k_nms_scan<<<1, 32, 0, stream>>>(Pbox, Pder, suprow, out);
}